// MacroTopologyGNN_7851200217222
// MI455X (gfx1250) — compile-verified
//
#include <hip/hip_runtime.h>
#include <hip/hip_bf16.h>
#include <math.h>

#define HID 128
#define NGRAPH 32
#define GN_EPS 1e-5f

typedef __attribute__((ext_vector_type(2))) float v2f;
typedef __attribute__((ext_vector_type(8))) float v8f;

// ---------------------------------------------------------------- zero ----
__global__ void k_zero(float* __restrict__ p, size_t n) {
    size_t i = (size_t)blockIdx.x * blockDim.x + threadIdx.x;
    size_t stride = (size_t)gridDim.x * blockDim.x;
    for (; i < n; i += stride) p[i] = 0.0f;
}

// --------------------------------------------- edge scatter (mean aggr) ----
// one thread per (edge, feature); f==0 thread also counts degree
__global__ void k_scatter(const float* __restrict__ x,
                          const int* __restrict__ src,
                          const int* __restrict__ dst,
                          float* __restrict__ msg,
                          float* __restrict__ deg,
                          int E) {
    int idx = blockIdx.x * blockDim.x + threadIdx.x;   // E*128 <= 102.4M < 2^31
    if (idx >= E * HID) return;
    int e = idx >> 7;
    int f = idx & (HID - 1);
    int s = src[e];
    int d = dst[e];
    atomicAdd(&msg[(size_t)d * HID + f], x[(size_t)s * HID + f]);
    if (f == 0) atomicAdd(&deg[d], 1.0f);
}

// ------------------------------------------------- fused dual WMMA GEMM ----
// pre = (msg/deg) @ Wl + x @ Wr + bl      (f32 WMMA 16x16x4, K=128)
// one wave -> one 16x16 tile; 8 waves/block cover the 8 column tiles.
__global__ __launch_bounds__(256) void k_gemm(const float* __restrict__ msg,
                                              const float* __restrict__ deg,
                                              const float* __restrict__ x,
                                              const float* __restrict__ Wl,
                                              const float* __restrict__ bl,
                                              const float* __restrict__ Wr,
                                              float* __restrict__ pre,
                                              int nrows) {
    const int lane  = threadIdx.x & 31;
    const int wave  = threadIdx.x >> 5;          // 0..7 = N-tile
    const int mtile = blockIdx.x;
    const int lrow  = lane & 15;
    const int khalf = lane >> 4;                 // 0 or 1 -> K pairs (0,1)/(2,3)
    int row = mtile * 16 + lrow;
    if (row >= nrows) row = nrows - 1;           // keep EXEC all-ones for WMMA
    const int col = wave * 16 + lrow;

    const float invd = 1.0f / fmaxf(deg[row], 1.0f);
    const float* __restrict__ arow = msg + (size_t)row * HID;
    const float* __restrict__ xrow = x   + (size_t)row * HID;

    v8f acc = {0.f, 0.f, 0.f, 0.f, 0.f, 0.f, 0.f, 0.f};

    // pass 1: agg @ Wl   (fold 1/deg into A operand)
    for (int k = 0; k < HID; k += 4) {
        const int kk = k + 2 * khalf;
        v2f a = *(const v2f*)(arow + kk);
        a = a * invd;
        v2f b;
        b.x = Wl[kk * HID + col];
        b.y = Wl[(kk + 1) * HID + col];
        acc = __builtin_amdgcn_wmma_f32_16x16x4_f32(false, a, false, b,
                                                    (short)0, acc, false, false);
    }
    // pass 2: x @ Wr   (same accumulator)
    for (int k = 0; k < HID; k += 4) {
        const int kk = k + 2 * khalf;
        v2f a = *(const v2f*)(xrow + kk);
        v2f b;
        b.x = Wr[kk * HID + col];
        b.y = Wr[(kk + 1) * HID + col];
        acc = __builtin_amdgcn_wmma_f32_16x16x4_f32(false, a, false, b,
                                                    (short)0, acc, false, false);
    }

    const float bias = bl[col];
#pragma unroll
    for (int v = 0; v < 8; ++v) {
        const int r = mtile * 16 + v + 8 * khalf;   // D layout: M = v + 8*(lane>>4)
        if (r < nrows) pre[(size_t)r * HID + col] = acc[v] + bias;
    }
}

// --------------------------------------- per-graph sums (batch is sorted) --
// blockDim = 128 (one thread per feature); each block scans 256 nodes,
// keeping a running sum and flushing one atomic per graph boundary.
__global__ __launch_bounds__(128) void k_gnsum(const float* __restrict__ pre,
                                               const int* __restrict__ batch,
                                               float* __restrict__ gsum,
                                               float* __restrict__ gcnt,
                                               int n) {
    const int f = threadIdx.x;
    const int start = blockIdx.x * 256;
    int end = start + 256; if (end > n) end = n;
    if (start >= n) return;
    int curg = batch[start];
    float run = 0.0f, cnt = 0.0f;
    for (int i = start; i < end; ++i) {
        const int g = batch[i];
        if (g != curg) {
            atomicAdd(&gsum[curg * HID + f], run);
            if (f == 0) atomicAdd(&gcnt[curg], cnt);
            run = 0.0f; cnt = 0.0f; curg = g;
        }
        run += pre[(size_t)i * HID + f];
        cnt += 1.0f;
    }
    atomicAdd(&gsum[curg * HID + f], run);
    if (f == 0) atomicAdd(&gcnt[curg], cnt);
}

__global__ __launch_bounds__(128) void k_gnvar(const float* __restrict__ pre,
                                               const int* __restrict__ batch,
                                               const float* __restrict__ gsum,
                                               const float* __restrict__ gcnt,
                                               const float* __restrict__ alpha,
                                               float* __restrict__ vsum,
                                               int n) {
    const int f = threadIdx.x;
    const int start = blockIdx.x * 256;
    int end = start + 256; if (end > n) end = n;
    if (start >= n) return;
    const float a = alpha[f];
    int curg = batch[start];
    float am = a * (gsum[curg * HID + f] / fmaxf(gcnt[curg], 1.0f));
    float run = 0.0f;
    for (int i = start; i < end; ++i) {
        const int g = batch[i];
        if (g != curg) {
            atomicAdd(&vsum[curg * HID + f], run);
            run = 0.0f; curg = g;
            am = a * (gsum[curg * HID + f] / fmaxf(gcnt[curg], 1.0f));
        }
        const float s = pre[(size_t)i * HID + f] - am;
        run += s * s;
    }
    atomicAdd(&vsum[curg * HID + f], run);
}

// ----------------------------------- normalize + exact GELU + residual ----
__global__ void k_final(const float* __restrict__ pre,
                        const int* __restrict__ batch,
                        const float* __restrict__ gsum,
                        const float* __restrict__ gcnt,
                        const float* __restrict__ vsum,
                        const float* __restrict__ alpha,
                        const float* __restrict__ gw,
                        const float* __restrict__ gb,
                        const float* __restrict__ x,
                        float* __restrict__ out,
                        size_t total) {
    size_t t = (size_t)blockIdx.x * blockDim.x + threadIdx.x;
    size_t stride = (size_t)gridDim.x * blockDim.x;
    for (; t < total; t += stride) {
        const size_t i = t >> 7;
        const int f = (int)(t & (HID - 1));
        const int g = batch[i];
        const float c = fmaxf(gcnt[g], 1.0f);
        const float mean = gsum[g * HID + f] / c;
        const float var  = vsum[g * HID + f] / c;
        const float sub  = pre[t] - alpha[f] * mean;
        const float nrm  = gw[f] * sub * rsqrtf(var + GN_EPS) + gb[f];
        const float gel  = 0.5f * nrm * (1.0f + erff(nrm * 0.70710678118654752f));
        out[t] = gel + x[t];
    }
}

// ---------------------------------------------------------------------------
extern "C" void kernel_launch(void* const* d_in, const int* in_sizes, int n_in,
                              void* d_out, int out_size, void* d_ws, size_t ws_size,
                              hipStream_t stream) {
    const float* x     = (const float*)d_in[0];
    const int*   ei    = (const int*)d_in[1];
    const int*   batch = (const int*)d_in[2];
    const float* Wl    = (const float*)d_in[3];
    const float* bl    = (const float*)d_in[4];
    const float* Wr    = (const float*)d_in[5];
    const float* gw    = (const float*)d_in[6];
    const float* gb    = (const float*)d_in[7];
    const float* galp  = (const float*)d_in[8];
    float* out = (float*)d_out;

    const int n = in_sizes[0] / HID;     // 50000 nodes
    const int E = in_sizes[1] / 2;       // 800000 edges
    const int* src = ei;
    const int* dst = ei + E;

    // workspace layout (floats)
    float* ws = (float*)d_ws;
    const size_t NH = (size_t)n * HID;
    float* msg  = ws;                    // [N, 128] neighbor sums
    float* pre  = ws + NH;               // [N, 128] pre-norm output
    float* tail = ws + 2 * NH;
    const size_t degPad = ((size_t)n + 63) & ~(size_t)63;
    float* deg  = tail;                  // [N] degree
    float* gsum = tail + degPad;         // [32, 128]
    float* gcnt = gsum + NGRAPH * HID;   // [32] (+pad)
    float* vsum = gcnt + 64;             // [32, 128]
    const size_t tailN = degPad + NGRAPH * HID + 64 + NGRAPH * HID;

    // 1) zero accumulators
    k_zero<<<2048, 256, 0, stream>>>(msg, NH);
    k_zero<<<256, 256, 0, stream>>>(tail, tailN);

    // 2) edge scatter-add (mean aggregation numerator + degree)
    {
        const int tot = E * HID;
        k_scatter<<<(tot + 255) / 256, 256, 0, stream>>>(x, src, dst, msg, deg, E);
    }

    // 3) fused dual GEMM via f32 WMMA: pre = (msg/deg)@Wl + x@Wr + bl
    k_gemm<<<(n + 15) / 16, 256, 0, stream>>>(msg, deg, x, Wl, bl, Wr, pre, n);

    // 4) per-graph feature sums + counts
    k_gnsum<<<(n + 255) / 256, 128, 0, stream>>>(pre, batch, gsum, gcnt, n);

    // 5) per-graph variance of (pre - alpha*mean)
    k_gnvar<<<(n + 255) / 256, 128, 0, stream>>>(pre, batch, gsum, gcnt, galp, vsum, n);

    // 6) GraphNorm affine + exact GELU + residual
    k_final<<<4096, 256, 0, stream>>>(pre, batch, gsum, gcnt, vsum, galp, gw, gb, x,
                                      out, NH);
}